// HeteroTimeGAT_21337397526698
// MI455X (gfx1250) — compile-verified
//
#include <hip/hip_runtime.h>
#include <hip/hip_bf16.h>
#include <math.h>

typedef __attribute__((ext_vector_type(2))) float v2f;
typedef __attribute__((ext_vector_type(8))) float v8f;

#define EMB 64
#define K3  192   // 3*EMB

// ---------------------------------------------------------------------------
// Kernel 1: per-edge attention scores via V_WMMA_F32_16X16X4_F32.
// One wave = 16 edges. A-fragment (16x4 f32): lanes 0-15 hold K={kk,kk+1},
// lanes 16-31 hold K={kk+2,kk+3} for row M=lane&15 -> one float2 load/lane.
// B = W replicated across all 16 columns (from LDS), so D[m][n] == score[m]
// for every n; lane 0 (M=0-7) and lane 16 (M=8-15) write the results.
// ---------------------------------------------------------------------------
__global__ void __launch_bounds__(256) score_kernel(
    const float* __restrict__ xr, const float* __restrict__ xt,
    const float* __restrict__ Wr, const float* __restrict__ br,
    const float* __restrict__ Wt, const float* __restrict__ bt,
    float* __restrict__ relScores, float* __restrict__ timeScores, int E)
{
    __shared__ float sW[2][K3];
    if (threadIdx.x < K3) {
        sW[0][threadIdx.x] = Wr[threadIdx.x];
        sW[1][threadIdx.x] = Wt[threadIdx.x];
    }
    __syncthreads();

    const int wave = threadIdx.x >> 5;
    const int lane = threadIdx.x & 31;
    const int half = lane >> 4;   // K sub-pair selector
    const int row  = lane & 15;   // M index

    const int e0 = blockIdx.x * 128 + wave * 16;
    int e = e0 + row;
    if (e >= E) e = E - 1;        // clamp: keep EXEC all-ones for WMMA
    const float* pr = xr + (size_t)e * K3 + 2 * half;
    const float* pt = xt + (size_t)e * K3 + 2 * half;

    v8f accR = {};
    v8f accT = {};
#pragma unroll 4
    for (int kk = 0; kk < K3; kk += 4) {
        v2f a = *(const v2f*)(pr + kk);   // 8B aligned
        v2f c = *(const v2f*)(pt + kk);
        v2f bR, bT;
        bR.x = sW[0][kk + 2 * half]; bR.y = sW[0][kk + 2 * half + 1];
        bT.x = sW[1][kk + 2 * half]; bT.y = sW[1][kk + 2 * half + 1];
        accR = __builtin_amdgcn_wmma_f32_16x16x4_f32(
            false, a, false, bR, (short)0, accR, false, false);
        accT = __builtin_amdgcn_wmma_f32_16x16x4_f32(
            false, c, false, bT, (short)0, accT, false, false);
    }

    const float b0 = br[0], b1 = bt[0];
    if ((lane & 15) == 0) {       // lane 0 -> M=0..7, lane 16 -> M=8..15
        const int base = e0 + (half ? 8 : 0);
#pragma unroll
        for (int i = 0; i < 8; ++i) {
            if (base + i < E) {
                relScores[base + i]  = accR[i] + b0;
                timeScores[base + i] = accT[i] + b1;
            }
        }
    }
}

// ---------------------------------------------------------------------------
// Kernel 2: per-node segment max & sum(exp). row_idx is sorted, so each node
// binary-searches its [start,end) edge range. Fully deterministic (no float
// atomics), trivially cheap vs the streaming passes.
// ---------------------------------------------------------------------------
__global__ void node_stats_kernel(
    const long long* __restrict__ rows,
    const float* __restrict__ relScores, const float* __restrict__ timeScores,
    float* __restrict__ relMax, float* __restrict__ relSum,
    float* __restrict__ timeMax, float* __restrict__ timeSum,
    int E, int N)
{
    int n = blockIdx.x * blockDim.x + threadIdx.x;
    if (n >= N) return;

    long long key = n;
    int lo = 0, hi = E;
    while (lo < hi) { int mid = (lo + hi) >> 1; if (rows[mid] < key) lo = mid + 1; else hi = mid; }
    const int start = lo;
    hi = E; key = n + 1;
    while (lo < hi) { int mid = (lo + hi) >> 1; if (rows[mid] < key) lo = mid + 1; else hi = mid; }
    const int end = lo;

    float mR = -INFINITY, mT = -INFINITY;
    for (int k = start; k < end; ++k) {
        mR = fmaxf(mR, relScores[k]);
        mT = fmaxf(mT, timeScores[k]);
    }
    float sR = 0.f, sT = 0.f;
    for (int k = start; k < end; ++k) {
        sR += expf(relScores[k] - mR);
        sT += expf(timeScores[k] - mT);
    }
    relMax[n] = mR; relSum[n] = sR;
    timeMax[n] = mT; timeSum[n] = sT;
}

// ---------------------------------------------------------------------------
// Kernel 3: scores -> attention weights, in place (per-edge independent).
// ---------------------------------------------------------------------------
__global__ void attn_kernel(
    const long long* __restrict__ rows,
    float* __restrict__ relScores, float* __restrict__ timeScores,
    const float* __restrict__ relMax, const float* __restrict__ relSum,
    const float* __restrict__ timeMax, const float* __restrict__ timeSum, int E)
{
    int e = blockIdx.x * blockDim.x + threadIdx.x;
    if (e >= E) return;
    int r = (int)rows[e];
    relScores[e]  = expf(relScores[e]  - relMax[r])  / relSum[r];
    timeScores[e] = expf(timeScores[e] - timeMax[r]) / timeSum[r];
}

// ---------------------------------------------------------------------------
// Kernel 4: out_rel = neighs * relAttn, out_time = neighs * timeAttn.
// float4 vectorized; reads neighs once, writes both output halves.
// ---------------------------------------------------------------------------
__global__ void __launch_bounds__(256) scatter_kernel(
    const float4* __restrict__ neighs4,
    const float* __restrict__ relAttn, const float* __restrict__ timeAttn,
    float4* __restrict__ out, size_t total4)
{
    size_t i = (size_t)blockIdx.x * blockDim.x + threadIdx.x;
    if (i >= total4) return;
    float4 nv = neighs4[i];
    size_t e = i >> 4;                 // (i*4)/64
    float aR = relAttn[e], aT = timeAttn[e];
    float4 o1 = { nv.x * aR, nv.y * aR, nv.z * aR, nv.w * aR };
    float4 o2 = { nv.x * aT, nv.y * aT, nv.z * aT, nv.w * aT };
    out[i] = o1;
    out[total4 + i] = o2;
}

extern "C" void kernel_launch(void* const* d_in, const int* in_sizes, int n_in,
                              void* d_out, int out_size, void* d_ws, size_t ws_size,
                              hipStream_t stream)
{
    const float*     xr     = (const float*)d_in[0];
    const float*     xt     = (const float*)d_in[1];
    /* d_in[2] = selfs: unused by the reference computation */
    const float*     neighs = (const float*)d_in[3];
    const long long* rows   = (const long long*)d_in[4];
    const float*     Wr     = (const float*)d_in[5];
    const float*     br     = (const float*)d_in[6];
    const float*     Wt     = (const float*)d_in[7];
    const float*     bt     = (const float*)d_in[8];

    const int E = in_sizes[4];           // row_idx element count
    const int N = in_sizes[2] / EMB;     // selfs rows

    // Workspace layout (floats): relScores[E] timeScores[E] relMax[N] relSum[N] timeMax[N] timeSum[N]
    float* ws         = (float*)d_ws;
    float* relScores  = ws;
    float* timeScores = ws + (size_t)E;
    float* relMax     = ws + 2 * (size_t)E;
    float* relSum     = relMax + N;
    float* timeMax    = relSum + N;
    float* timeSum    = timeMax + N;

    const int blocks1 = (E + 127) / 128;   // 128 edges per 256-thread block (8 waves x 16)
    score_kernel<<<blocks1, 256, 0, stream>>>(xr, xt, Wr, br, Wt, bt,
                                              relScores, timeScores, E);
    node_stats_kernel<<<(N + 255) / 256, 256, 0, stream>>>(
        rows, relScores, timeScores, relMax, relSum, timeMax, timeSum, E, N);
    attn_kernel<<<(E + 255) / 256, 256, 0, stream>>>(
        rows, relScores, timeScores, relMax, relSum, timeMax, timeSum, E);

    const size_t total4 = ((size_t)E * EMB) / 4;
    scatter_kernel<<<(unsigned)((total4 + 255) / 256), 256, 0, stream>>>(
        (const float4*)neighs, relScores, timeScores, (float4*)d_out, total4);
}